// Net_10969346474792
// MI455X (gfx1250) — compile-verified
//
#include <hip/hip_runtime.h>
#include <hip/hip_bf16.h>
#include <math.h>

typedef float v2f __attribute__((ext_vector_type(2)));
typedef float v8f __attribute__((ext_vector_type(8)));

#define F_IN  256
#define HID   128
#define FOUT  64
#define NGRAPH 512

// ---------------------------------------------------------------------------
// Utility: fill / zero
// ---------------------------------------------------------------------------
__global__ void k_fill(float* __restrict__ p, long n, float v) {
    long i = (long)blockIdx.x * blockDim.x + threadIdx.x;
    if (i < n) p[i] = v;
}

// ---------------------------------------------------------------------------
// Degree accumulation: deg[dst] += 1  (deg pre-filled with 1.0 for self-loop)
// ---------------------------------------------------------------------------
__global__ void k_deg(const int* __restrict__ dst, float* __restrict__ deg, int E) {
    int e = blockIdx.x * blockDim.x + threadIdx.x;
    if (e < E) atomicAdd(&deg[dst[e]], 1.0f);
}

__global__ void k_rsqrt(float* __restrict__ d, int n) {
    int i = blockIdx.x * blockDim.x + threadIdx.x;
    if (i < n) d[i] = rsqrtf(d[i]);
}

// ---------------------------------------------------------------------------
// WMMA f32 GEMM:  OUT[M,128] = A[M,K] x B[K,128]
// B given as two matrices: cols [0,SPLIT) from B0 (width SPLIT), cols
// [SPLIT,128) from B1 (width 128-SPLIT). 256-thread block = 8 waves; wave w
// owns M-tile (blockIdx.x*8 + w). W is staged through LDS in 64-row K-chunks
// (32 KB), shared by all 8 waves.
//
// LDS layout is ROW-PAIR INTERLEAVED: sW[p*256 + n*2 + r] = W[2p+r][n].
// A B-fragment {W[kk][n], W[kk+1][n]} (kk = k0+2*half, even) is then one
// contiguous float2 -> single ds_load_b64, no repack movs; all tile offsets
// are DS immediate offsets off one base VGPR.
//
// V_WMMA_F32_16X16X4_F32 layouts (wave32):
//   A 16x4 : lane l: m=l&15, half=l>>4 ; v0=A[m][k0+2*half], v1=A[m][k0+2*half+1]
//   B 4x16 : lane l: n=l&15, half=l>>4 ; v0=B[k0+2*half][n], v1=B[k0+2*half+1][n]
//   C 16x16: VGPR v holds row m=v+8*half, col n=l&15
// ---------------------------------------------------------------------------
template<int K, int SPLIT>
__global__ __launch_bounds__(256) void k_gemm_wmma(
    const float* __restrict__ A, const float* __restrict__ B0,
    const float* __restrict__ B1, int M, float* __restrict__ OUT)
{
    constexpr int KC = 64;                       // K-chunk rows staged in LDS
    __shared__ float sW[(KC / 2) * 256];         // 32 KB, pair-interleaved

    const int tid  = threadIdx.x;
    const int wave = tid >> 5;
    const int lane = tid & 31;
    const int half = lane >> 4;
    const int m    = lane & 15;
    const int row0 = (blockIdx.x * 8 + wave) * 16;

    // clamp A row for safety (keeps EXEC all-1s through the WMMA loop)
    int arow = row0 + m; if (arow >= M) arow = M - 1;
    const float* __restrict__ aPtr = A + (size_t)arow * K + 2 * half;

    v8f acc[8];
#pragma unroll
    for (int t = 0; t < 8; ++t) acc[t] = (v8f){0,0,0,0,0,0,0,0};

    for (int kc = 0; kc < K; kc += KC) {
        __syncthreads();
        // Stage rows [kc, kc+KC) pair-interleaved. Thread i handles row-pair
        // p = i>>6 and column pair c = (i&63)*2: two float2 global reads,
        // one float4 LDS write {W[2p][c], W[2p+1][c], W[2p][c+1], W[2p+1][c+1]}.
#pragma unroll
        for (int i = tid; i < (KC / 2) * 64; i += 256) {
            const int p  = i >> 6;
            const int c  = (i & 63) * 2;
            const int k0g = kc + 2 * p;
            float2 r0, r1;
            if constexpr (SPLIT < 128) {
                if (c < SPLIT) {
                    r0 = *(const float2*)(B0 + (size_t)k0g * SPLIT + c);
                    r1 = *(const float2*)(B0 + (size_t)(k0g + 1) * SPLIT + c);
                } else {
                    constexpr int W1W = 128 - SPLIT;
                    r0 = *(const float2*)(B1 + (size_t)k0g * W1W + (c - SPLIT));
                    r1 = *(const float2*)(B1 + (size_t)(k0g + 1) * W1W + (c - SPLIT));
                }
            } else {
                r0 = *(const float2*)(B0 + (size_t)k0g * 128 + c);
                r1 = *(const float2*)(B0 + (size_t)(k0g + 1) * 128 + c);
            }
            float4 v; v.x = r0.x; v.y = r1.x; v.z = r0.y; v.w = r1.y;
            *(float4*)(sW + p * 256 + c * 2) = v;
        }
        __syncthreads();

        // base for this lane's B fragments: pair index (k0/2 + half), col m
        const float* __restrict__ wbase = sW + half * 256 + m * 2;
#pragma unroll 4
        for (int k0 = 0; k0 < KC; k0 += 4) {
            v2f a;
            {
                const float2 av = *(const float2*)(aPtr + kc + k0);
                a.x = av.x; a.y = av.y;
            }
            const float* __restrict__ wrow = wbase + (k0 / 2) * 256;
#pragma unroll
            for (int t = 0; t < 8; ++t) {
                const v2f b = *(const v2f*)(wrow + t * 32);   // ds_load_b64
                acc[t] = __builtin_amdgcn_wmma_f32_16x16x4_f32(
                    false, a, false, b, (short)0, acc[t], false, false);
            }
        }
    }

#pragma unroll
    for (int t = 0; t < 8; ++t) {
#pragma unroll
        for (int v = 0; v < 8; ++v) {
            const int row = row0 + half * 8 + v;
            if (row < M)
                OUT[(size_t)row * 128 + t * 16 + m] = acc[t][v];
        }
    }
}

// ---------------------------------------------------------------------------
// Edge message/aggregate:  AGG[dst] += H[src] * (dis[src]*dis[dst])
// 32 threads per edge, one float4 feature chunk each (width 128).
// ---------------------------------------------------------------------------
__global__ void k_msg128(const int* __restrict__ src, const int* __restrict__ dst,
                         const float* __restrict__ dis, const float* __restrict__ H,
                         float* __restrict__ AGG, long total)
{
    long idx = (long)blockIdx.x * blockDim.x + threadIdx.x;
    if (idx >= total) return;
    const int e = (int)(idx >> 5);
    const int c = (int)(idx & 31) * 4;
    const int s = src[e], d = dst[e];
    const float w = dis[s] * dis[d];
    const float4 h = *(const float4*)(H + (size_t)s * 128 + c);
    float* p = AGG + (size_t)d * 128 + c;
    atomicAdd(p + 0, h.x * w);
    atomicAdd(p + 1, h.y * w);
    atomicAdd(p + 2, h.z * w);
    atomicAdd(p + 3, h.w * w);
}

// ---------------------------------------------------------------------------
// Layer-1 finalize:  A <- relu(AGG + H1*dis^2 + b1)   (A holds H1 on entry)
// ---------------------------------------------------------------------------
__global__ void k_fin1(float* __restrict__ A, const float* __restrict__ AGG,
                       const float* __restrict__ dis, const float* __restrict__ b1,
                       long total)
{
    long idx = (long)blockIdx.x * blockDim.x + threadIdx.x;
    if (idx >= total) return;
    const int n = (int)(idx >> 5);
    const int c = (int)(idx & 31) * 4;
    const float d2 = dis[n] * dis[n];
    float4 h  = *(const float4*)(A   + (size_t)n * 128 + c);
    float4 g  = *(const float4*)(AGG + (size_t)n * 128 + c);
    float4 bb = *(const float4*)(b1 + c);
    float4 r;
    r.x = fmaxf(g.x + h.x * d2 + bb.x, 0.0f);
    r.y = fmaxf(g.y + h.y * d2 + bb.y, 0.0f);
    r.z = fmaxf(g.z + h.z * d2 + bb.z, 0.0f);
    r.w = fmaxf(g.w + h.w * d2 + bb.w, 0.0f);
    *(float4*)(A + (size_t)n * 128 + c) = r;
}

// ---------------------------------------------------------------------------
// z finalize + pooled sums:
//   mean = AGG[:, :64] + H34[:, :64]*dis^2 + b3
//   lstd = AGG[:,64: ] + H34[:,64: ]*dis^2 + b4
//   z = mean + noise*exp(lstd);   sums[batch[n]] += z;  counts[batch[n]] += 1
// ---------------------------------------------------------------------------
__global__ void k_finz(const float* __restrict__ AGG, const float* __restrict__ H34,
                       const float* __restrict__ dis, const float* __restrict__ b3,
                       const float* __restrict__ b4, const float* __restrict__ noise,
                       const int* __restrict__ batch, float* __restrict__ sums,
                       float* __restrict__ counts, long total)
{
    long idx = (long)blockIdx.x * blockDim.x + threadIdx.x;
    if (idx >= total) return;
    const int n = (int)(idx >> 4);
    const int c = (int)(idx & 15) * 4;
    const float d2 = dis[n] * dis[n];
    const float4 am = *(const float4*)(AGG + (size_t)n * 128 + c);
    const float4 hm = *(const float4*)(H34 + (size_t)n * 128 + c);
    const float4 as = *(const float4*)(AGG + (size_t)n * 128 + 64 + c);
    const float4 hs = *(const float4*)(H34 + (size_t)n * 128 + 64 + c);
    const float4 nz = *(const float4*)(noise + (size_t)n * 64 + c);
    const float4 c3 = *(const float4*)(b3 + c);
    const float4 c4 = *(const float4*)(b4 + c);
    float4 z;
    z.x = (am.x + hm.x * d2 + c3.x) + nz.x * expf(as.x + hs.x * d2 + c4.x);
    z.y = (am.y + hm.y * d2 + c3.y) + nz.y * expf(as.y + hs.y * d2 + c4.y);
    z.z = (am.z + hm.z * d2 + c3.z) + nz.z * expf(as.z + hs.z * d2 + c4.z);
    z.w = (am.w + hm.w * d2 + c3.w) + nz.w * expf(as.w + hs.w * d2 + c4.w);
    const int g = batch[n];
    float* p = sums + (size_t)g * 64 + c;
    atomicAdd(p + 0, z.x);
    atomicAdd(p + 1, z.y);
    atomicAdd(p + 2, z.z);
    atomicAdd(p + 3, z.w);
    if ((idx & 15) == 0) atomicAdd(&counts[g], 1.0f);
}

// ---------------------------------------------------------------------------
// Head: pooled = sums/max(counts,1); logits = pooled@Wfc + bfc; log_softmax
// ---------------------------------------------------------------------------
__global__ void k_head(const float* __restrict__ sums, const float* __restrict__ counts,
                       const float* __restrict__ Wfc, const float* __restrict__ bfc,
                       float* __restrict__ out)
{
    int g = blockIdx.x * blockDim.x + threadIdx.x;
    if (g >= NGRAPH) return;
    const float inv = 1.0f / fmaxf(counts[g], 1.0f);
    float lg0 = bfc[0], lg1 = bfc[1], lg2 = bfc[2], lg3 = bfc[3];
#pragma unroll 4
    for (int f = 0; f < 64; ++f) {
        const float p = sums[(size_t)g * 64 + f] * inv;
        lg0 += p * Wfc[f * 4 + 0];
        lg1 += p * Wfc[f * 4 + 1];
        lg2 += p * Wfc[f * 4 + 2];
        lg3 += p * Wfc[f * 4 + 3];
    }
    float mx = fmaxf(fmaxf(lg0, lg1), fmaxf(lg2, lg3));
    float se = expf(lg0 - mx) + expf(lg1 - mx) + expf(lg2 - mx) + expf(lg3 - mx);
    float lse = mx + logf(se);
    out[g * 4 + 0] = lg0 - lse;
    out[g * 4 + 1] = lg1 - lse;
    out[g * 4 + 2] = lg2 - lse;
    out[g * 4 + 3] = lg3 - lse;
}

// ---------------------------------------------------------------------------
extern "C" void kernel_launch(void* const* d_in, const int* in_sizes, int n_in,
                              void* d_out, int out_size, void* d_ws, size_t ws_size,
                              hipStream_t stream) {
    const float* x     = (const float*)d_in[0];
    const int*   eidx  = (const int*)  d_in[1];
    const int*   batch = (const int*)  d_in[2];
    const float* W1    = (const float*)d_in[3];
    const float* b1    = (const float*)d_in[4];
    const float* W3    = (const float*)d_in[5];
    const float* b3    = (const float*)d_in[6];
    const float* W4    = (const float*)d_in[7];
    const float* b4    = (const float*)d_in[8];
    const float* Wfc   = (const float*)d_in[9];
    const float* bfc   = (const float*)d_in[10];
    const float* noise = (const float*)d_in[11];
    float* out = (float*)d_out;

    const int N = in_sizes[0] / F_IN;
    const int E = in_sizes[1] / 2;
    const int* src = eidx;
    const int* dst = eidx + E;

    // workspace layout (256B aligned)
    char* ws = (char*)d_ws;
    size_t off = 0;
    auto take = [&](size_t bytes) {
        size_t o = off;
        off = (off + bytes + 255) & ~(size_t)255;
        return o;
    };
    float* dis    = (float*)(ws + take((size_t)N * 4));          // deg -> dis in place
    float* bufA   = (float*)(ws + take((size_t)N * 128 * 4));    // H1 -> Hrelu -> AGG34
    float* bufB   = (float*)(ws + take((size_t)N * 128 * 4));    // AGG1 -> H34
    float* sums   = (float*)(ws + take((size_t)NGRAPH * 64 * 4));
    float* counts = (float*)(ws + take((size_t)NGRAPH * 4));

    const int TB = 256;
    const long nh  = (long)N * 128;           // node*feature elements (width 128)
    const long nh4 = (long)N * 32;            // float4 chunks (width 128)
    const long ez4 = (long)E * 32;            // edge float4 chunks (width 128)
    const long nz4 = (long)N * 16;            // node float4 chunks (width 64)
    auto blocks = [&](long n) { return (unsigned)((n + TB - 1) / TB); };
    const unsigned gemm_blocks = (unsigned)((N + 127) / 128);    // 8 M-tiles / block

    // --- degree / normalization ---
    k_fill<<<blocks(N), TB, 0, stream>>>(dis, N, 1.0f);
    k_deg<<<blocks(E), TB, 0, stream>>>(dst, dis, E);
    k_rsqrt<<<blocks(N), TB, 0, stream>>>(dis, N);

    // --- conv1: H1 = x @ W1 (WMMA), aggregate, relu ---
    k_fill<<<blocks(nh), TB, 0, stream>>>(bufB, nh, 0.0f);       // zero AGG1
    k_gemm_wmma<F_IN, 128><<<gemm_blocks, 256, 0, stream>>>(x, W1, W1, N, bufA);
    k_msg128<<<blocks(ez4), TB, 0, stream>>>(src, dst, dis, bufA, bufB, ez4);
    k_fin1<<<blocks(nh4), TB, 0, stream>>>(bufA, bufB, dis, b1, nh4);

    // --- conv2+conv3 fused: H34 = Hrelu @ [W3|W4] (WMMA), aggregate ---
    k_gemm_wmma<HID, 64><<<gemm_blocks, 256, 0, stream>>>(bufA, W3, W4, N, bufB);
    k_fill<<<blocks(nh), TB, 0, stream>>>(bufA, nh, 0.0f);       // zero AGG34
    k_msg128<<<blocks(ez4), TB, 0, stream>>>(src, dst, dis, bufB, bufA, ez4);

    // --- z + pooled sums ---
    k_fill<<<blocks(NGRAPH * 64), TB, 0, stream>>>(sums, NGRAPH * 64, 0.0f);
    k_fill<<<blocks(NGRAPH), TB, 0, stream>>>(counts, NGRAPH, 0.0f);
    k_finz<<<blocks(nz4), TB, 0, stream>>>(bufA, bufB, dis, b3, b4, noise, batch,
                                           sums, counts, nz4);

    // --- head ---
    k_head<<<(NGRAPH + TB - 1) / TB, TB, 0, stream>>>(sums, counts, Wfc, bfc, out);
}